// DiceLoss_12704513262234
// MI455X (gfx1250) — compile-verified
//
#include <hip/hip_runtime.h>
#include <hip/hip_bf16.h>
#include <stdint.h>

// ---------------------------------------------------------------------------
// DiceLoss for predict/target [2,4,64,256,256] fp32.
// Bandwidth-bound: 256 MB read once -> ~11.5us floor @ 23.3 TB/s.
// Strategy: 512 workgroups (one per contiguous 64K-float (b,o,d) slice),
// 256 threads each, double-buffered async global->LDS b128 streaming,
// sigmoid via v_exp_f32/v_rcp_f32, wave reduction via chained
// v_wmma_f32_16x16x4_f32 against an all-ones B matrix.
// ---------------------------------------------------------------------------

typedef float v2f __attribute__((ext_vector_type(2)));
typedef float v8f __attribute__((ext_vector_type(8)));

#define NSLICES 512          // B*O*D = 2*4*64
#define HW      65536        // 256*256 elements per slice
#define THREADS 256
#define ITERS   64           // (HW/4 float4s) / THREADS

// Low 32 bits of a generic pointer into LDS are the LDS byte offset
__device__ __forceinline__ uint32_t lds_addr_of(const void* p) {
  return (uint32_t)(uintptr_t)p;
}

// GVS-mode async copy: copies 16 bytes from global address sbase plus byte offset voff to the LDS address in ldsb, ASYNCcnt++.
__device__ __forceinline__ void async_ld_b128(uint32_t ldsb, uint32_t voff,
                                              const void* sbase) {
  asm volatile("global_load_async_to_lds_b128 %0, %1, %2"
               :: "v"(ldsb), "v"(voff), "s"(sbase)
               : "memory");
}

#define WAIT_ASYNC(N) asm volatile("s_wait_asynccnt " #N ::: "memory")

// Full wave32 sum broadcast to all lanes using two chained WMMAs.
// Stage 1: A = {x_lane, 0} (A[m][0]=x_m, A[m][2]=x_{m+16}), B = ones
//   -> D[m][n] = x_m + x_{m+16} = S_m  (D vgpr j: lanes0-15 = S_j, 16-31 = S_{j+8})
//   in-lane sum of 8 D vgprs -> lanes0-15 = sum(S_0..7), lanes16-31 = sum(S_8..15)
// Stage 2: same trick -> every element = total sum.
// All-ones B is correct under any operand layout. EXEC must be all ones
// (called with no divergence, full 256-thread blocks).
__device__ __forceinline__ float wave_reduce_sum(float x) {
  v2f ones; ones[0] = 1.0f; ones[1] = 1.0f;
  v2f a;    a[0] = x;       a[1] = 0.0f;
  v8f c = {};
  c = __builtin_amdgcn_wmma_f32_16x16x4_f32(false, a, false, ones,
                                            (short)0, c, false, false);
  float y = ((c[0] + c[1]) + (c[2] + c[3])) + ((c[4] + c[5]) + (c[6] + c[7]));
  v2f a2; a2[0] = y; a2[1] = 0.0f;
  v8f c2 = {};
  c2 = __builtin_amdgcn_wmma_f32_16x16x4_f32(false, a2, false, ones,
                                             (short)0, c2, false, false);
  return c2[0];
}

__global__ void __launch_bounds__(THREADS)
dice_slice_kernel(const float* __restrict__ predict,
                  const float* __restrict__ target,
                  float* __restrict__ ws) {
  __shared__ float4 bufP[2][THREADS];   // 8 KB
  __shared__ float4 bufT[2][THREADS];   // 8 KB
  __shared__ float  wred[3][8];

  const int s   = blockIdx.x;           // slice id (b*O*D + o*D + d)
  const int tid = threadIdx.x;
  const uint32_t sliceBytes = (uint32_t)s * (HW * 4u);

  // Prime the pipeline: stage 0.
  {
    uint32_t off = sliceBytes + (uint32_t)tid * 16u;
    async_ld_b128(lds_addr_of(&bufP[0][tid]), off, predict);
    async_ld_b128(lds_addr_of(&bufT[0][tid]), off, target);
  }

  float accN = 0.0f;   // sum p*t
  float accP = 0.0f;   // sum p
  float accT = 0.0f;   // sum t
  const float LOG2E = 1.4426950408889634f;

  for (int it = 0; it < ITERS; ++it) {
    const int buf = it & 1;
    if (it + 1 < ITERS) {
      uint32_t off = sliceBytes + ((uint32_t)(it + 1) * THREADS + tid) * 16u;
      const int nb = (it + 1) & 1;
      async_ld_b128(lds_addr_of(&bufP[nb][tid]), off, predict);
      async_ld_b128(lds_addr_of(&bufT[nb][tid]), off, target);
      WAIT_ASYNC(2);      // 4 in flight; oldest 2 (current stage) done in-order
    } else {
      WAIT_ASYNC(0);      // drain last stage
    }

    float4 p4 = bufP[buf][tid];   // ds_load_b128, own-lane slot only
    float4 t4 = bufT[buf][tid];
    float px[4] = {p4.x, p4.y, p4.z, p4.w};
    float tx[4] = {t4.x, t4.y, t4.z, t4.w};
#pragma unroll
    for (int k = 0; k < 4; ++k) {
      float e  = __builtin_amdgcn_exp2f(-px[k] * LOG2E);  // exp(-x)
      float sg = __builtin_amdgcn_rcpf(1.0f + e);         // sigmoid(x)
      accN = fmaf(sg, tx[k], accN);
      accP += sg;
      accT += tx[k];
    }
  }

  // Wave-level reduction on the matrix pipe (uniform control flow -> EXEC all 1s).
  float rn = wave_reduce_sum(accN);
  float rp = wave_reduce_sum(accP);
  float rt = wave_reduce_sum(accT);

  const int wave = tid >> 5;
  const int lane = tid & 31;
  if (lane == 0) {
    wred[0][wave] = rn;
    wred[1][wave] = rp;
    wred[2][wave] = rt;
  }
  __syncthreads();

  if (tid == 0) {
    float N = 0.f, P = 0.f, T = 0.f;
#pragma unroll
    for (int w = 0; w < 8; ++w) { N += wred[0][w]; P += wred[1][w]; T += wred[2][w]; }
    float den  = P + T + 1.0f;                 // SMOOTH = 1.0
    float dice = 1.0f - 2.0f * N / den;
    float t0   = target[(size_t)s * HW];       // tf[..., 0]
    float valid = (t0 != -1.0f) ? 1.0f : 0.0f;
    ws[s]           = dice;
    ws[NSLICES + s] = valid;
  }
}

// Masked per-(b,o) average over d, then mean over the 8 (b,o) pairs.
__global__ void __launch_bounds__(NSLICES)
dice_finalize_kernel(const float* __restrict__ ws, float* __restrict__ out) {
  __shared__ float gnum[8];
  __shared__ float gden[8];
  const int t = threadIdx.x;
  if (t < 8) { gnum[t] = 0.0f; gden[t] = 0.0f; }
  __syncthreads();
  if (t < NSLICES) {
    float dice  = ws[t];
    float valid = ws[NSLICES + t];
    int g = t >> 6;                     // 64 d-slices per (b,o)
    atomicAdd(&gnum[g], dice * valid);  // ds_add_f32
    atomicAdd(&gden[g], valid);
  }
  __syncthreads();
  if (t == 0) {
    float acc = 0.0f;
#pragma unroll
    for (int g = 0; g < 8; ++g) acc += gnum[g] / gden[g];
    out[0] = acc * 0.125f;
  }
}

extern "C" void kernel_launch(void* const* d_in, const int* in_sizes, int n_in,
                              void* d_out, int out_size, void* d_ws, size_t ws_size,
                              hipStream_t stream) {
  const float* predict = (const float*)d_in[0];
  const float* target  = (const float*)d_in[1];
  float* ws  = (float*)d_ws;     // 1024 floats: [dice x512 | valid x512]
  float* out = (float*)d_out;    // 1 float

  dice_slice_kernel<<<NSLICES, THREADS, 0, stream>>>(predict, target, ws);
  dice_finalize_kernel<<<1, NSLICES, 0, stream>>>(ws, out);
}